// MandelbrotKernel_3083786518612
// MI455X (gfx1250) — compile-verified
//
#include <hip/hip_runtime.h>

// Mandelbrot escape-time counts, f32, branchless, 2 independent chains/thread.
//
// Reference semantics: z0 = c; at step k in [0,256]: if active && |z|^2>=4 ->
// counts = k, deactivate; then z = z^2 + c if still active. Step 256 can only
// write counts=256 where counts already is 256 (init), so steps 0..255
// suffice. With z frozen at first escape (finite, |z|^2>=4), the predicate
// |z|^2 < 4 is monotonically false after escape, so
//   counts = sum_{k=0}^{255} (|z_k|^2 < 4)
// equals the first-escape step (or 256 if never escaped). Fully branchless:
// one v_cmp + cndmask-freeze per pixel per iteration.
//
// Roofline: 201 MB of traffic (~8.6 us @ 23.3 TB/s) vs ~5e10 VALU-lane-ops --
// pure f32 VALU throughput problem. ~24% of the window is interior, so
// P(wave32 has an interior lane) ~ 99.98%: every wave runs all 256 iterations
// and per-wave early exit is pure overhead. Two independent pixel chains per
// thread feed VOPD dual-issue; __f*_rn blocks FMA contraction so roundings
// match the reference's separate mul/add ops (p+p == (2*zr)*zi bit-exactly).

#define MB_MAX_ITERS 256

__global__ __launch_bounds__(256) void mandelbrot_f32_kernel(
    const float* __restrict__ c_real,
    const float* __restrict__ c_imag,
    float* __restrict__ out,
    int n, int half)
{
    const int t  = blockIdx.x * blockDim.x + threadIdx.x;
    const int i0 = t;             // chain 0: first half of the image
    const int i1 = t + half;      // chain 1: second half of the image
    const bool v0 = (i0 < half);  // i0 < half <= n
    const bool v1 = (i1 < n);

    // Unconditional coalesced loads with clamped indices: no exec-mask
    // branches in the preamble. Out-of-range threads compute on a duplicated
    // pixel and simply don't store.
    const int j0 = v0 ? i0 : (n - 1);
    const int j1 = v1 ? i1 : (n - 1);
    const float cr0 = c_real[j0];
    const float ci0 = c_imag[j0];
    const float cr1 = c_real[j1];
    const float ci1 = c_imag[j1];

    float zr0 = cr0, zi0 = ci0, cnt0 = 0.0f;
    float zr1 = cr1, zi1 = ci1, cnt1 = 0.0f;

#pragma unroll 8
    for (int k = 0; k < MB_MAX_ITERS; ++k) {
        // chain 0
        float zr2_0 = __fmul_rn(zr0, zr0);
        float zi2_0 = __fmul_rn(zi0, zi0);
        float m0    = __fadd_rn(zr2_0, zi2_0);
        bool  a0    = (m0 < 4.0f);
        // chain 1 (independent -> VOPD dual-issue fodder)
        float zr2_1 = __fmul_rn(zr1, zr1);
        float zi2_1 = __fmul_rn(zi1, zi1);
        float m1    = __fadd_rn(zr2_1, zi2_1);
        bool  a1    = (m1 < 4.0f);

        cnt0 = __fadd_rn(cnt0, a0 ? 1.0f : 0.0f);
        cnt1 = __fadd_rn(cnt1, a1 ? 1.0f : 0.0f);

        // next_real = zr^2 - zi^2 + cr   (separate roundings, no FMA contract)
        float nr0 = __fadd_rn(__fsub_rn(zr2_0, zi2_0), cr0);
        float nr1 = __fadd_rn(__fsub_rn(zr2_1, zi2_1), cr1);
        // next_imag = 2*zr*zi + ci
        float p0  = __fmul_rn(zr0, zi0);
        float p1  = __fmul_rn(zr1, zi1);
        float ni0 = __fadd_rn(__fadd_rn(p0, p0), ci0);
        float ni1 = __fadd_rn(__fadd_rn(p1, p1), ci1);

        // Freeze z on escape (keeps m >= 4 forever, makes count trick exact).
        zr0 = a0 ? nr0 : zr0;
        zi0 = a0 ? ni0 : zi0;
        zr1 = a1 ? nr1 : zr1;
        zi1 = a1 ? ni1 : zi1;
    }

    if (v0) out[i0] = cnt0;
    if (v1) out[i1] = cnt1;
}

extern "C" void kernel_launch(void* const* d_in, const int* in_sizes, int n_in,
                              void* d_out, int out_size, void* d_ws, size_t ws_size,
                              hipStream_t stream) {
    (void)n_in; (void)out_size; (void)d_ws; (void)ws_size;

    const float* c_real = (const float*)d_in[0];
    const float* c_imag = (const float*)d_in[1];
    float*       out    = (float*)d_out;

    const int n    = in_sizes[0];      // 4096*4096 = 16,777,216
    const int half = (n + 1) / 2;      // 2 pixels per thread

    const int threads = 256;           // 8 wave32 per block
    const int blocks  = (half + threads - 1) / threads;

    mandelbrot_f32_kernel<<<blocks, threads, 0, stream>>>(c_real, c_imag, out, n, half);
}